// SpatialAttention3D_48490180772049
// MI455X (gfx1250) — compile-verified
//
#include <hip/hip_runtime.h>
#include <cmath>

#define BB 4
#define CC 64
#define DDD 32
#define HHH 64
#define WWW 64
#define DHW (DDD*HHH*WWW)   // 131072
#define HW  (HHH*WWW)       // 4096

typedef float v8f __attribute__((ext_vector_type(8)));
typedef float v2f __attribute__((ext_vector_type(2)));

#if defined(__has_builtin)
#  if __has_builtin(__builtin_amdgcn_global_load_async_to_lds_b32)
#    define USE_ASYNC_LDS 1
#  endif
#endif

// ---------------- Kernel 1: channel mean + max pooling ----------------
__global__ void sa3d_pool_kernel(const float* __restrict__ x,
                                 float* __restrict__ pooled) {
  int t = blockIdx.x * blockDim.x + threadIdx.x;   // 0 .. B*DHW-1
  if (t >= BB * DHW) return;
  int b   = t / DHW;
  int dhw = t - b * DHW;
  const float* p = x + (size_t)b * CC * DHW + dhw;
  float s = 0.f;
  float m = -INFINITY;
#pragma unroll 4
  for (int c = 0; c < CC; ++c) {
    float v = p[(size_t)c * DHW];
    int cp = (c + 2 < CC) ? (c + 2) : c;   // distance-2 channel prefetch
    __builtin_prefetch(p + (size_t)cp * DHW, 0, 0);
    s += v;
    m = fmaxf(m, v);
  }
  pooled[(size_t)(b * 2 + 0) * DHW + dhw] = s * (1.0f / CC);
  pooled[(size_t)(b * 2 + 1) * DHW + dhw] = m;
}

// ---------------- Kernel 2: 7x7x7 conv (2ch -> 1ch) + sigmoid via WMMA ----
// Block = 128 threads (4 waves); one (b, d, 16-row h-tile) per block, each
// wave owns a 16(h) x 16(w) output tile. Per (c,kd,kh) the W-direction conv
// is a 16x24 (input rows) x 24x16 (zero-padded banded weights) product done
// as six V_WMMA_F32_16X16X4_F32 chunks. Band zeros live in the LDS layout,
// so B loads are plain ds_load_b64 pairs -- no EXEC predication.
__global__ __launch_bounds__(128)
void sa3d_conv_att_kernel(const float* __restrict__ pooled,
                          const float* __restrict__ cw,
                          float* __restrict__ att) {
  __shared__ float sIn[7][22][73];   // kd planes, 22 halo rows, 72 cols (+1 pad)
  __shared__ float sWP[98][48];      // 98 padded banded weight rows, cols [0,40) valid

  int blk   = blockIdx.x;            // 0 .. B*D*4 - 1
  int htile = blk & 3;
  int bd    = blk >> 2;
  int d     = bd & (DDD - 1);
  int b     = bd >> 5;
  int h0    = htile * 16;

  int tid  = threadIdx.x;
  int lane = tid & 31;
  int wave = tid >> 5;
  int wt0  = wave * 16;              // this wave's w-tile origin
  int j    = lane & 15;              // N column (w) == M row (h) index
  int koff = (lane >> 4) * 2;        // per-lane K sub-offset (ISA A/B layout)

  // zero the padded weight rows, then drop the 7 taps at column 16
  for (int t = tid; t < 98 * 40; t += 128) sWP[t / 40][t % 40] = 0.f;
  __syncthreads();
  for (int t = tid; t < 686; t += 128) sWP[t / 7][16 + (t % 7)] = cw[t];

  v8f acc = {};

  for (int c = 0; c < 2; ++c) {
    __syncthreads();
    // stage zero-padded halo planes for this channel: kd 0..6, 22 rows, 72 cols
    for (int t = tid; t < 7 * 22 * 72; t += 128) {
      int kd = t / (22 * 72);
      int rr = t - kd * (22 * 72);
      int r  = rr / 72;
      int wl = rr - r * 72;
      int gd = d + kd - 3;
      int gh = h0 - 3 + r;
      int gw = wl - 3;
      bool inr = (gd >= 0 && gd < DDD && gh >= 0 && gh < HHH && gw >= 0 && gw < WWW);
#ifdef USE_ASYNC_LDS
      if (inr) {
        const float* gp =
            pooled + (size_t)((b * 2 + c) * DDD + gd) * HW + gh * WWW + gw;
        __builtin_amdgcn_global_load_async_to_lds_b32(
            (__attribute__((address_space(1))) int*)(void*)gp,
            (__attribute__((address_space(3))) int*)&sIn[kd][r][wl],
            0, 0);
      } else {
        sIn[kd][r][wl] = 0.f;
      }
#else
      float v = 0.f;
      if (inr)
        v = pooled[(size_t)((b * 2 + c) * DDD + gd) * HW + gh * WWW + gw];
      sIn[kd][r][wl] = v;
#endif
    }
#ifdef USE_ASYNC_LDS
#  if __has_builtin(__builtin_amdgcn_s_wait_asynccnt)
    __builtin_amdgcn_s_wait_asynccnt(0);
#  else
    asm volatile("s_wait_asynccnt 0" ::: "memory");
#  endif
#endif
    __syncthreads();

    for (int kd = 0; kd < 7; ++kd) {
      for (int kh = 0; kh < 7; ++kh) {
        const float* aBase = &sIn[kd][j + kh][wt0 + koff];
        const float* bBase = &sWP[c * 49 + kd * 7 + kh][16 - j + koff];
#pragma unroll
        for (int n = 0; n < 6; ++n) {
          v2f a, bb;
          a.x  = aBase[4 * n];
          a.y  = aBase[4 * n + 1];
          bb.x = bBase[4 * n];
          bb.y = bBase[4 * n + 1];
          acc = __builtin_amdgcn_wmma_f32_16x16x4_f32(
              false, a, false, bb, (short)0, acc, false, false);
        }
      }
    }
  }

  // D layout: VGPR r -> M = r (+8 for lanes 16-31), N = lane&15
  int    hi   = (lane >> 4) * 8;
  float* outb = att + (size_t)(b * DDD + d) * HW;
#pragma unroll
  for (int r = 0; r < 8; ++r) {
    int   h = h0 + r + hi;
    int   w = wt0 + j;
    float v = acc[r];
    outb[h * WWW + w] = 1.0f / (1.0f + __expf(-v));
  }
}

// ---------------- Kernel 3: out = x * att (broadcast over C), float4 -------
__global__ void sa3d_scale_kernel(const float* __restrict__ x,
                                  const float* __restrict__ att,
                                  float* __restrict__ out) {
  int t = blockIdx.x * blockDim.x + threadIdx.x;  // over 8,388,608 float4s
  const float4* x4 = (const float4*)x;
  const float4* a4 = (const float4*)att;
  float4*       o4 = (float4*)out;
  int b    = t >> 21;       // t / (C*DHW/4)
  int dhw4 = t & 32767;     // (4t % DHW)/4
  float4 xv = x4[t];
  float4 av = a4[b * 32768 + dhw4];
  float4 ov;
  ov.x = xv.x * av.x;
  ov.y = xv.y * av.y;
  ov.z = xv.z * av.z;
  ov.w = xv.w * av.w;
  o4[t] = ov;
}

extern "C" void kernel_launch(void* const* d_in, const int* in_sizes, int n_in,
                              void* d_out, int out_size, void* d_ws, size_t ws_size,
                              hipStream_t stream) {
  const float* x  = (const float*)d_in[0];   // [4,64,32,64,64] f32
  const float* cw = (const float*)d_in[1];   // [1,2,7,7,7]    f32
  float*       out = (float*)d_out;

  float* pooled = (float*)d_ws;                   // [B,2,D,H,W] -> 4 MiB
  float* attw   = pooled + (size_t)BB * 2 * DHW;  // [B,D,H,W]   -> 2 MiB

  sa3d_pool_kernel<<<(BB * DHW) / 256, 256, 0, stream>>>(x, pooled);
  sa3d_conv_att_kernel<<<BB * DDD * (HHH / 16), 128, 0, stream>>>(pooled, cw, attw);
  sa3d_scale_kernel<<<(BB * CC * DHW / 4) / 256, 256, 0, stream>>>(x, attw, out);
}